// customGAT_63290638074150
// MI455X (gfx1250) — compile-verified
//
#include <hip/hip_runtime.h>
#include <math.h>

typedef __attribute__((ext_vector_type(2))) float v2f;
typedef __attribute__((ext_vector_type(8))) float v8f;
typedef __attribute__((ext_vector_type(4))) unsigned int u32x4;
typedef __attribute__((ext_vector_type(8))) int i32x8;
typedef __attribute__((ext_vector_type(4))) int i32x4;

#define NEG_SLOPE 0.2f

__device__ __forceinline__ void atomic_max_f32(float* addr, float v) {
    // IEEE-754 trick: signed-int compare works for non-negative floats,
    // unsigned-min for negative floats.
    if (v >= 0.0f) atomicMax((int*)addr, __float_as_int(v));
    else           atomicMin((unsigned int*)addr, __float_as_uint(v));
}

__global__ void fill_f32(float* __restrict__ p, float v, long n) {
    long i = (long)blockIdx.x * blockDim.x + threadIdx.x;
    if (i < n) p[i] = v;
}

// C[M,Ncols] = act(A[M,K] + abias) @ B[K,Ncols], fp32 via V_WMMA_F32_16X16X4_F32.
// Block = 256 threads = 8 waves. Each wave: one 16x16 tile of C.
// The block's 16-wide B column slab (K x 16, 16KB) is DMA'd into LDS by the
// Tensor Data Mover (one tensor_load_to_lds issued by wave 0, TENSORcnt-tracked),
// then shared by all 8 waves via ds loads.
template <bool FUSE_BIAS_RELU>
__global__ __launch_bounds__(256) void gemm_f32_wmma(
    const float* __restrict__ A, const float* __restrict__ B, float* __restrict__ C,
    int M, int K, int Ncols, const float* __restrict__ abias)
{
    __shared__ float sB[256 * 16];           // K<=256 rows x 16 cols, row-major
    const int tid  = threadIdx.x;
    const int wave = tid >> 5;
    const int lane = tid & 31;
    const int n0   = blockIdx.x * 16;
    const int m0   = blockIdx.y * 128 + wave * 16;

    // --- TDM: DMA B[:, n0:n0+16] (tile 16 x K, row stride Ncols) into sB ---
    if (wave == 0) {
        unsigned long long ga = (unsigned long long)(uintptr_t)(B + (size_t)n0);
        unsigned int ldsOff = (unsigned int)(uintptr_t)sB;   // low 32 bits of flat LDS addr = LDS byte offset

        u32x4 g0;
        g0[0] = 1u;                                          // count=1 (valid user descriptor)
        g0[1] = ldsOff;                                      // lds_addr [63:32]
        g0[2] = (unsigned int)ga;                            // global_addr[31:0]  -> bits [95:64]
        g0[3] = (unsigned int)((ga >> 32) & 0x01FFFFFFu)     // global_addr[56:32] -> bits [120:96]
              | (2u << 30);                                  // type=2 ("image")  -> bits [127:126]

        i32x8 g1;
        g1[0] = (int)(2u << 16);                             // wg_mask=0, data_size=2 (4 bytes)
        g1[1] = (int)(((unsigned)Ncols & 0xFFFFu) << 16);    // tensor_dim0[15:0] at bits [63:48]
        g1[2] = (int)((((unsigned)Ncols >> 16) & 0xFFFFu)    // tensor_dim0[31:16]
              | (((unsigned)K & 0xFFFFu) << 16));            // tensor_dim1[15:0]
        g1[3] = (int)((((unsigned)K >> 16) & 0xFFFFu)        // tensor_dim1[31:16]
              | (16u << 16));                                // tile_dim0 = 16
        g1[4] = (int)((unsigned)K & 0xFFFFu);                // tile_dim1 = K, tile_dim2 = 0
        g1[5] = (int)(unsigned)Ncols;                        // tensor_dim0_stride[31:0] = Ncols
        g1[6] = 0;                                           // stride0[47:32]=0, stride1 lo=0
        g1[7] = 0;                                           // stride1 hi=0

        i32x4 gz4 = {0, 0, 0, 0};                            // groups 2/3 unused (2D tensor)
        i32x8 gz8 = {0, 0, 0, 0, 0, 0, 0, 0};
        __builtin_amdgcn_tensor_load_to_lds(g0, g1, gz4, gz4, gz8, 0);
        __builtin_amdgcn_s_wait_tensorcnt(0);
    }
    __syncthreads();

    const int halfHi = (lane >> 4) & 1;      // 0: lanes 0-15, 1: lanes 16-31
    const int mrow   = m0 + (lane & 15);     // A-matrix row held by this lane
    const int rowA   = mrow < M ? mrow : M - 1;  // clamp (store is guarded)
    const float* Arow = A + (size_t)rowA * K;
    const int nc = lane & 15;                // C column within tile

    v8f acc = {0.f, 0.f, 0.f, 0.f, 0.f, 0.f, 0.f, 0.f};
    for (int k = 0; k < K; k += 4) {
        if ((k & 63) == 0) __builtin_prefetch(Arow + k + 64, 0, 3);
        const int kk = k + (halfHi << 1);    // lanes 16-31 hold K=k+2,k+3
        v2f a;
        a.x = Arow[kk];
        a.y = Arow[kk + 1];
        if constexpr (FUSE_BIAS_RELU) {      // layer-2: relu(agg + b1) fused on the fly
            a.x = fmaxf(a.x + abias[kk], 0.f);
            a.y = fmaxf(a.y + abias[kk + 1], 0.f);
        }
        v2f b;                               // B: row kk / kk+1 at column nc
        b.x = sB[kk * 16 + nc];
        b.y = sB[(kk + 1) * 16 + nc];
        acc = __builtin_amdgcn_wmma_f32_16x16x4_f32(
            /*neg_a=*/false, a, /*neg_b=*/false, b,
            /*c_mod=*/(short)0, acc, /*reuse_a=*/false, /*reuse_b=*/false);
    }

    #pragma unroll
    for (int v = 0; v < 8; ++v) {            // VGPR v: row m0+v (lo half) / m0+8+v (hi half)
        int row = m0 + v + (halfHi << 3);
        if (row < M) C[(size_t)row * Ncols + n0 + nc] = acc[v];
    }
}

// Per-node attention halves: as[n] = h[n,:].a_src, ad[n] = h[n,:].a_dst. Wave per node.
__global__ __launch_bounds__(256) void node_alpha(
    const float* __restrict__ H, const float* __restrict__ avs, const float* __restrict__ avd,
    float* __restrict__ as, float* __restrict__ ad, int M, int C)
{
    int gw   = (int)(((long)blockIdx.x * blockDim.x + threadIdx.x) >> 5);
    int lane = threadIdx.x & 31;
    if (gw >= M) return;
    const float* h = H + (size_t)gw * C;
    float ss = 0.f, sd = 0.f;
    for (int c = lane; c < C; c += 32) {
        float hv = h[c];
        ss += hv * avs[c];
        sd += hv * avd[c];
    }
    #pragma unroll
    for (int off = 16; off > 0; off >>= 1) {
        ss += __shfl_down(ss, off, 32);
        sd += __shfl_down(sd, off, 32);
    }
    if (lane == 0) { as[gw] = ss; ad[gw] = sd; }
}

__device__ __forceinline__ void edge_nodes(const int* __restrict__ ei, int E, int e,
                                           int& src, int& dst) {
    if (e < E) { src = ei[e]; dst = ei[E + e]; }
    else       { src = dst = e - E; }        // appended self loops
}

__global__ void edge_max(const int* __restrict__ ei, int E, int ET,
                         const float* __restrict__ as, const float* __restrict__ ad,
                         float* __restrict__ m)
{
    int e = blockIdx.x * blockDim.x + threadIdx.x;
    if (e >= ET) return;
    int src, dst; edge_nodes(ei, E, e, src, dst);
    float v = as[src] + ad[dst];
    v = v > 0.f ? v : v * NEG_SLOPE;
    atomic_max_f32(&m[dst], v);
}

__global__ void edge_expsum(const int* __restrict__ ei, int E, int ET,
                            const float* __restrict__ as, const float* __restrict__ ad,
                            const float* __restrict__ m, float* __restrict__ denom,
                            float* __restrict__ exb)
{
    int e = blockIdx.x * blockDim.x + threadIdx.x;
    if (e >= ET) return;
    int src, dst; edge_nodes(ei, E, e, src, dst);
    float v = as[src] + ad[dst];
    v = v > 0.f ? v : v * NEG_SLOPE;
    float ex = __expf(v - m[dst]);
    exb[e] = ex;
    atomicAdd(&denom[dst], ex);
}

// Wave per edge: out[dst,:] += (ex/denom[dst]) * H[src,:]. Lane-strided => coalesced.
__global__ __launch_bounds__(256) void edge_scatter(
    const int* __restrict__ ei, int E, int ET,
    const float* __restrict__ H, const float* __restrict__ exb,
    const float* __restrict__ denom, float* __restrict__ out, int C)
{
    int gw   = (int)(((long)blockIdx.x * blockDim.x + threadIdx.x) >> 5);
    int lane = threadIdx.x & 31;
    if (gw >= ET) return;
    int src, dst; edge_nodes(ei, E, gw, src, dst);
    float alpha = exb[gw] / denom[dst];
    const float* hs = H + (size_t)src * C;
    float* od = out + (size_t)dst * C;
    for (int c = lane; c < C; c += 32)
        atomicAdd(&od[c], alpha * hs[c]);
}

__global__ void add_bias(float* __restrict__ p, const float* __restrict__ b, long n, int C) {
    long i = (long)blockIdx.x * blockDim.x + threadIdx.x;
    if (i < n) p[i] += b[i & (C - 1)];       // C is a power of two (256 / 128)
}

extern "C" void kernel_launch(void* const* d_in, const int* in_sizes, int n_in,
                              void* d_out, int out_size, void* d_ws, size_t ws_size,
                              hipStream_t stream) {
    const float* x   = (const float*)d_in[0];
    const int*   ei  = (const int*)  d_in[1];
    const float* W1  = (const float*)d_in[2];
    const float* a1s = (const float*)d_in[3];
    const float* a1d = (const float*)d_in[4];
    const float* b1  = (const float*)d_in[5];
    const float* W2  = (const float*)d_in[6];
    const float* a2s = (const float*)d_in[7];
    const float* a2d = (const float*)d_in[8];
    const float* b2  = (const float*)d_in[9];
    float* out = (float*)d_out;

    const int IN = 256, HID = 256, OUTC = 128;
    const int N  = in_sizes[0] / IN;
    const int E  = in_sizes[1] / 2;
    const int ET = E + N;

    // Workspace carve-out (~133 MB).
    char* w = (char*)d_ws;
    auto carve = [&](size_t bytes) -> float* {
        float* p = (float*)w;
        w += (bytes + 255) & ~(size_t)255;
        return p;
    };
    float* H1   = carve((size_t)N * HID  * 4);
    float* AGG1 = carve((size_t)N * HID  * 4);
    float* H2   = carve((size_t)N * OUTC * 4);
    float* as1  = carve((size_t)N * 4);
    float* ad1  = carve((size_t)N * 4);
    float* m1   = carve((size_t)N * 4);
    float* dn1  = carve((size_t)N * 4);
    float* as2  = carve((size_t)N * 4);
    float* ad2  = carve((size_t)N * 4);
    float* m2   = carve((size_t)N * 4);
    float* dn2  = carve((size_t)N * 4);
    float* exb  = carve((size_t)ET * 4);

    auto blk1d = [](long n, int bs) { return (unsigned)((n + bs - 1) / bs); };

    // ---- init (must re-init every call; buffers are not re-poisoned) ----
    fill_f32<<<blk1d((long)N * HID, 256), 256, 0, stream>>>(AGG1, 0.f, (long)N * HID);
    fill_f32<<<blk1d((long)N * OUTC, 256), 256, 0, stream>>>(out, 0.f, (long)N * OUTC);
    fill_f32<<<blk1d(N, 256), 256, 0, stream>>>(m1, -INFINITY, N);
    fill_f32<<<blk1d(N, 256), 256, 0, stream>>>(dn1, 0.f, N);
    fill_f32<<<blk1d(N, 256), 256, 0, stream>>>(m2, -INFINITY, N);
    fill_f32<<<blk1d(N, 256), 256, 0, stream>>>(dn2, 0.f, N);

    // ---- layer 1 ----
    {
        dim3 g(HID / 16, (N + 127) / 128);
        gemm_f32_wmma<false><<<g, 256, 0, stream>>>(x, W1, H1, N, IN, HID, nullptr);
    }
    node_alpha<<<blk1d((long)N * 32, 256), 256, 0, stream>>>(H1, a1s, a1d, as1, ad1, N, HID);
    edge_max   <<<blk1d(ET, 256), 256, 0, stream>>>(ei, E, ET, as1, ad1, m1);
    edge_expsum<<<blk1d(ET, 256), 256, 0, stream>>>(ei, E, ET, as1, ad1, m1, dn1, exb);
    edge_scatter<<<blk1d((long)ET * 32, 256), 256, 0, stream>>>(ei, E, ET, H1, exb, dn1, AGG1, HID);

    // ---- layer 2 (bias b1 + ReLU fused into the A operand of the GEMM) ----
    {
        dim3 g(OUTC / 16, (N + 127) / 128);
        gemm_f32_wmma<true><<<g, 256, 0, stream>>>(AGG1, W2, H2, N, HID, OUTC, b1);
    }
    node_alpha<<<blk1d((long)N * 32, 256), 256, 0, stream>>>(H2, a2s, a2d, as2, ad2, N, OUTC);
    edge_max   <<<blk1d(ET, 256), 256, 0, stream>>>(ei, E, ET, as2, ad2, m2);
    edge_expsum<<<blk1d(ET, 256), 256, 0, stream>>>(ei, E, ET, as2, ad2, m2, dn2, exb);
    edge_scatter<<<blk1d((long)ET * 32, 256), 256, 0, stream>>>(ei, E, ET, H2, exb, dn2, out, OUTC);

    add_bias<<<blk1d((long)N * OUTC, 256), 256, 0, stream>>>(out, b2, (long)N * OUTC, OUTC);
}